// MechanicsFunctions_59167469469778
// MI455X (gfx1250) — compile-verified
//
#include <hip/hip_runtime.h>
#include <stdint.h>

// Element Hessian for neo-Hookean energy, closed form:
// H[a,i,b,k] = vol * [ MU*s_ab*delta_ik + LAM*h[3a+i]*h[3b+k] + (MU-LAM*logJ)*h[3a+k]*h[3b+i] ]
// Phase 2 builds the rank-structured terms on the WMMA pipe: 4x rank-1
// v_wmma_f32_16x16x4_f32 outer products per element (K-order robust: identical
// vector in all K slots of A and B), combined fully branch-free per D slot.

#define EPB 256          // elements per block == threads per block
#define CORE_STRIDE 33   // floats per element in LDS core (30 used)
#define MU_  1.0f
#define LAM_ 1.0f

typedef __attribute__((ext_vector_type(2))) float v2f;
typedef __attribute__((ext_vector_type(8))) float v8f;

__global__ __launch_bounds__(EPB)
void hess_kernel(const float* __restrict__ U,      // (N,3)
                 const float* __restrict__ G,      // shape_grads (E,1,4,3)
                 const float* __restrict__ vols,   // (E,1)
                 const int*   __restrict__ conns,  // (E,4)
                 float* __restrict__ out,          // (E,4,3,4,3)
                 int E)
{
    __shared__ float lds[EPB * 12 + EPB + EPB * CORE_STRIDE];
    float* Gt = lds;                 // staged shape_grads tile: EPB*12 floats
    float* Vt = lds + EPB * 12;      // staged vols tile: EPB floats
    float* Co = lds + EPB * 13;      // per-element core: EPB*CORE_STRIDE floats

    const int tid   = threadIdx.x;
    const int e0    = blockIdx.x * EPB;
    const int valid = (E - e0 < EPB) ? (E - e0) : EPB;

    // ---------------- Phase 0: async stage G and vols tiles into LDS ----------------
    {
        const float* src  = G + (size_t)e0 * 12;
        uint32_t     dst0 = (uint32_t)(uintptr_t)Gt;   // low 32 bits of flat addr == LDS offset
        const int    nch  = valid * 3;                 // 16-byte chunks (valid*48 B)
        for (int c = tid; c < nch; c += EPB) {
            const float* gp = src + c * 4;
            uint32_t     d  = dst0 + (uint32_t)c * 16u;
            asm volatile("global_load_async_to_lds_b128 %0, %1, off"
                         :: "v"(d), "v"(gp) : "memory");
        }
        const float* vsrc  = vols + e0;
        uint32_t     vdst0 = (uint32_t)(uintptr_t)Vt;
        const int    nv4   = valid >> 2;
        for (int c = tid; c < nv4; c += EPB) {
            const float* vp = vsrc + c * 4;
            uint32_t     d  = vdst0 + (uint32_t)c * 16u;
            asm volatile("global_load_async_to_lds_b128 %0, %1, off"
                         :: "v"(d), "v"(vp) : "memory");
        }
        const int rem = valid & 3;
        if (tid < rem) {
            const float* vp = vsrc + (nv4 * 4 + tid);
            uint32_t     d  = vdst0 + (uint32_t)(nv4 * 4 + tid) * 4u;
            asm volatile("global_load_async_to_lds_b32 %0, %1, off"
                         :: "v"(d), "v"(vp) : "memory");
        }
    }
    asm volatile("s_wait_asynccnt 0x0" ::: "memory");
    __syncthreads();

    // ---------------- Phase 1: one element per lane -> 30-float core in LDS ----------------
    if (tid < valid) {
        const int4 cn = ((const int4*)conns)[e0 + tid];
        const int  ca[4] = {cn.x, cn.y, cn.z, cn.w};

        float dm[4][3];
#pragma unroll
        for (int a = 0; a < 4; ++a) {
            const float* up = U + (size_t)ca[a] * 3;
            dm[a][0] = up[0]; dm[a][1] = up[1]; dm[a][2] = up[2];
        }
        float g[4][3];
#pragma unroll
        for (int a = 0; a < 4; ++a)
#pragma unroll
            for (int j = 0; j < 3; ++j)
                g[a][j] = Gt[tid * 12 + a * 3 + j];

        float F[3][3];
#pragma unroll
        for (int i = 0; i < 3; ++i)
#pragma unroll
            for (int j = 0; j < 3; ++j) {
                float s = (i == j) ? 1.0f : 0.0f;
#pragma unroll
                for (int a = 0; a < 4; ++a) s += dm[a][i] * g[a][j];
                F[i][j] = s;
            }

        const float c00 = F[1][1]*F[2][2] - F[1][2]*F[2][1];
        const float c01 = F[1][2]*F[2][0] - F[1][0]*F[2][2];
        const float c02 = F[1][0]*F[2][1] - F[1][1]*F[2][0];
        const float det = F[0][0]*c00 + F[0][1]*c01 + F[0][2]*c02;
        const float inv = 1.0f / det;

        float Fi[3][3];
        Fi[0][0] = c00 * inv;
        Fi[1][0] = c01 * inv;
        Fi[2][0] = c02 * inv;
        Fi[0][1] = (F[0][2]*F[2][1] - F[0][1]*F[2][2]) * inv;
        Fi[1][1] = (F[0][0]*F[2][2] - F[0][2]*F[2][0]) * inv;
        Fi[2][1] = (F[0][1]*F[2][0] - F[0][0]*F[2][1]) * inv;
        Fi[0][2] = (F[0][1]*F[1][2] - F[0][2]*F[1][1]) * inv;
        Fi[1][2] = (F[0][2]*F[1][0] - F[0][0]*F[1][2]) * inv;
        Fi[2][2] = (F[0][0]*F[1][1] - F[0][1]*F[1][0]) * inv;

        const float logJ = logf(det);
        const float vol  = Vt[tid];

        float* cw = Co + tid * CORE_STRIDE;
#pragma unroll
        for (int a = 0; a < 4; ++a)
#pragma unroll
            for (int i = 0; i < 3; ++i)
                cw[3*a + i] = g[a][0]*Fi[0][i] + g[a][1]*Fi[1][i] + g[a][2]*Fi[2][i];
#pragma unroll
        for (int a = 0; a < 4; ++a)
#pragma unroll
            for (int b = 0; b < 4; ++b)
                cw[12 + a*4 + b] = vol * MU_ *
                    (g[a][0]*g[b][0] + g[a][1]*g[b][1] + g[a][2]*g[b][2]);
        cw[28] = vol * LAM_;                  // c1
        cw[29] = vol * (MU_ - LAM_ * logJ);   // c2
    }
    __syncthreads();

    // ---------------- Phase 2: WMMA rank-1 assembly, one element per wave-iter ----
    const int lane  = tid & 31;
    const int wv    = tid >> 5;
    const int elBeg = wv * 32;
    int       elEnd = elBeg + 32;
    if (elEnd > valid) elEnd = valid;

    // ---- per-lane loop invariants ----
    const int   m     = lane & 15;           // A-row / B-col / D-col index
    const bool  mv    = (m < 12);
    const float mMask = mv ? 1.0f : 0.0f;
    const int   mcl   = mv ? m : 0;          // clamped for safe LDS reads
    const int   ma3   = (mcl / 3) * 3;
    const int   mi0   = mcl - ma3;
    const int   rot1  = ma3 + (mi0 + 1) % 3;
    const int   rot2  = ma3 + (mi0 + 2) % 3;

    const int   hi    = lane >> 4;           // D row-half selector
    const bool  hiB   = (hi != 0);
    const float hf    = hiB ? 1.0f : 0.0f;
    const int   Ncl   = mv ? m : 11;
    const int   nb    = Ncl / 3;
    const int   nk    = Ncl - 3 * nb;

    // per-slot invariant 0/1 masks: which T feeds slot r; diagonal split by half
    float st0[8], st1[8], st2[8], dA[8], dB[8];
#pragma unroll
    for (int r = 0; r < 8; ++r) {
        const int miR = hiB ? ((r + 2) % 3) : (r % 3);   // (r + 8*hi) % 3
        const int t   = (nk - miR + 3) % 3;
        st0[r] = (t == 0) ? 1.0f : 0.0f;
        st1[r] = (t == 1) ? 1.0f : 0.0f;
        st2[r] = (t == 2) ? 1.0f : 0.0f;
        const float dlr = (miR == nk) ? 1.0f : 0.0f;
        dA[r] = dlr * (1.0f - hf);           // selects t1vs[r/3]      (hi==0)
        dB[r] = dlr * hf;                    // selects t1vs[(r+8)/3]  (hi==1)
    }

    // per-lane output base: out[e*144 + (8*hi + r)*12 + N], store offsets r*48B
    float* obase = out + (size_t)(e0 + elBeg) * 144 + (hi * 96 + m);
    const float* cbase = Co + elBeg * CORE_STRIDE;

    for (int el = elBeg; el < elEnd; ++el, obase += 144, cbase += CORE_STRIDE) {
        const float* c  = cbase;
        const float  c1 = c[28];
        const float  c2 = c[29];
        const float  q1 = 0.25f * c1;
        const float  q2 = 0.25f * c2;

        const float v0 = mMask * c[mcl];
        const float w1 = mMask * c[rot1];
        const float w2 = mMask * c[rot2];

        // Gram row for this lane's column-block (unconditional loads)
        const float t1vs[4] = { c[12 + nb], c[16 + nb], c[20 + nb], c[24 + nb] };

        // All K slots identical => result independent of hardware K ordering.
        v2f Av  = {v0, v0};
        v2f A1  = {w1, w1};
        v2f A2  = {w2, w2};
        v2f Bc1 = {q1 * v0, q1 * v0};
        v2f Bt0 = {q2 * v0, q2 * v0};
        v2f Bt1 = {q2 * w2, q2 * w2};
        v2f Bt2 = {q2 * w1, q2 * w1};

        v8f D2 = {}, T0 = {}, T1 = {}, T2 = {};
        D2 = __builtin_amdgcn_wmma_f32_16x16x4_f32(false, Av, false, Bc1, (short)0, D2, false, false);
        T0 = __builtin_amdgcn_wmma_f32_16x16x4_f32(false, Av, false, Bt0, (short)0, T0, false, false);
        T1 = __builtin_amdgcn_wmma_f32_16x16x4_f32(false, A1, false, Bt1, (short)0, T1, false, false);
        T2 = __builtin_amdgcn_wmma_f32_16x16x4_f32(false, A2, false, Bt2, (short)0, T2, false, false);

        if (mv) {                      // N < 12 : rows M = 8*hi + r, r = 0..3 always valid
#pragma unroll
            for (int r = 0; r < 4; ++r) {
                float v = D2[r] + st0[r]*T0[r] + st1[r]*T1[r] + st2[r]*T2[r]
                        + dA[r]*t1vs[r / 3] + dB[r]*t1vs[(r + 8) / 3];
                obase[r * 12] = v;
            }
            if (!hiB) {                // rows 4..7 exist only in low half
#pragma unroll
                for (int r = 4; r < 8; ++r) {
                    float v = D2[r] + st0[r]*T0[r] + st1[r]*T1[r] + st2[r]*T2[r]
                            + dA[r]*t1vs[r / 3];
                    obase[r * 12] = v;
                }
            }
        }
    }
}

extern "C" void kernel_launch(void* const* d_in, const int* in_sizes, int n_in,
                              void* d_out, int out_size, void* d_ws, size_t ws_size,
                              hipStream_t stream) {
    // setup_inputs order: U, coords, state, shapes, shape_grads, vols, conns
    const float* U     = (const float*)d_in[0];
    const float* G     = (const float*)d_in[4];
    const float* vols  = (const float*)d_in[5];
    const int*   conns = (const int*)d_in[6];
    float*       out   = (float*)d_out;

    const int E      = in_sizes[5];            // vols is (E,1)
    const int blocks = (E + EPB - 1) / EPB;

    hipLaunchKernelGGL(hess_kernel, dim3(blocks), dim3(EPB), 0, stream,
                       U, G, vols, conns, out, E);
}